// TrfNetL1Sum_74955769249871
// MI455X (gfx1250) — compile-verified
//
#include <hip/hip_runtime.h>
#include <hip/hip_bf16.h>
#include <math.h>

// ---------------------------------------------------------------------------
// Sizes (compile-time constants from the reference)
// ---------------------------------------------------------------------------
#define N_NODES 10000
#define N_EDGES 100000
#define F_NODE  128
#define F_EDGE  16
#define HEADS   8
#define OUT_CH  128
#define HC      1024   // HEADS*OUT_CH

typedef __attribute__((ext_vector_type(2))) float    v2f;
typedef __attribute__((ext_vector_type(8))) float    v8f;
typedef __attribute__((ext_vector_type(4))) unsigned v4u;
typedef __attribute__((ext_vector_type(8))) unsigned v8u;

// ---------------------------------------------------------------------------
// Workspace layout (float offsets)
// ---------------------------------------------------------------------------
#define Q_OFF      0u
#define K_OFF      10240000u
#define V_OFF      20480000u
#define SKIP_OFF   30720000u
#define ALPHA_OFF  32000000u
#define MENC_OFF   32800000u   // unsigned[80000]
#define S_OFF      32880000u
#define OUTU_OFF   32960000u
#define POOL_OFF   43200000u
#define ZERO_N4    2600032     // (80000+80000+10240000+128)/4

// ---------------------------------------------------------------------------
// TDM stage: copy We (16x1024 f32 = 64KB, contiguous) into LDS with one
// tensor_load_to_lds issued by wave 0. D# per CDNA5 ISA ch.8:
//   group0: [1:0]=count=1 | [63:32]=lds_addr | [120:64]=global_addr | [127:126]=type=2
//   group1: data_size=2(4B), tensor_dim0=16384, tensor_dim1=1,
//           tile_dim0=16384, tile_dim1=1, tensor_dim0_stride=16384
// Completion tracked on TENSORcnt; other waves gated by the workgroup barrier.
// ---------------------------------------------------------------------------
__device__ __forceinline__ void stage_We_tdm(const float* __restrict__ We,
                                             float* WeS, int tid)
{
    if (tid < 32) {
        const unsigned lds = (unsigned)(unsigned long long)
            (__attribute__((address_space(3))) const float*)WeS;
        const unsigned long long ga = (unsigned long long)We;
        v4u g0;
        g0[0] = 1u;                                   // count = 1 valid descriptor
        g0[1] = lds;                                  // lds_addr
        g0[2] = (unsigned)ga;                         // global_addr[31:0]
        g0[3] = (unsigned)((ga >> 32) & 0x01FFFFFFull) | 0x80000000u; // addr[56:32] | type=2
        v8u g1;
        g1[0] = 0x00020000u;                          // data_size = 2 (4 bytes)
        g1[1] = 0x40000000u;                          // tensor_dim0[15:0]=16384 in [31:16]
        g1[2] = 0x00010000u;                          // tensor_dim0[31:16]=0, tensor_dim1[15:0]=1
        g1[3] = 0x40000000u;                          // tile_dim0 = 16384 in [31:16]
        g1[4] = 1u;                                   // tile_dim1 = 1
        g1[5] = 16384u;                               // tensor_dim0_stride[31:0]
        g1[6] = 0u;
        g1[7] = 0u;
        asm volatile("tensor_load_to_lds %0, %1" :: "s"(g0), "s"(g1) : "memory");
        __builtin_amdgcn_s_wait_tensorcnt(0);
    }
    __syncthreads();
}

// ---------------------------------------------------------------------------
// K0: zero fill (mEnc, s, out_unnorm, pooled)
// ---------------------------------------------------------------------------
__global__ void k0_zero(float4* __restrict__ p, int n4) {
    int i = blockIdx.x * blockDim.x + threadIdx.x;
    if (i < n4) p[i] = make_float4(0.f, 0.f, 0.f, 0.f);
}

// ---------------------------------------------------------------------------
// K1: fused QKV + skip GEMM with V_WMMA_F32_16X16X4_F32
//   grid.x = 625 (16-row tiles of x), grid.y = 50 (64-col tiles:
//   0..15 -> q, 16..31 -> k, 32..47 -> v, 48..49 -> skip). 128 thr = 4 waves,
//   each wave owns one 16x16 f32 accumulator tile.
// ---------------------------------------------------------------------------
__global__ __launch_bounds__(128)
void k1_qkvskip_gemm(const float* __restrict__ x,
                     const float* __restrict__ Wq, const float* __restrict__ bq,
                     const float* __restrict__ Wk, const float* __restrict__ bk,
                     const float* __restrict__ Wv, const float* __restrict__ bv,
                     const float* __restrict__ Wskip, const float* __restrict__ bskip,
                     float* __restrict__ q, float* __restrict__ k,
                     float* __restrict__ v, float* __restrict__ skip)
{
    __shared__ float xs[16 * 128];   //  8 KB : x tile (row-major, contiguous)
    __shared__ float wt[128 * 64];   // 32 KB : W tile

    const int tid  = threadIdx.x;
    const int row0 = blockIdx.x * 16;
    const int ty   = blockIdx.y;

    const float* W; const float* bias; float* out; int ldW, ldOut, n0;
    if (ty < 16)      { W = Wq;    bias = bq;    out = q;    ldW = HC;     ldOut = HC;     n0 = ty * 64; }
    else if (ty < 32) { W = Wk;    bias = bk;    out = k;    ldW = HC;     ldOut = HC;     n0 = (ty - 16) * 64; }
    else if (ty < 48) { W = Wv;    bias = bv;    out = v;    ldW = HC;     ldOut = HC;     n0 = (ty - 32) * 64; }
    else              { W = Wskip; bias = bskip; out = skip; ldW = OUT_CH; ldOut = OUT_CH; n0 = (ty - 48) * 64; }

    // stage x tile: 16 rows x 128 f32 are contiguous in memory
    {
        const float4* src = (const float4*)(x + (size_t)row0 * F_NODE);
        float4* dst = (float4*)xs;
        for (int i = tid; i < 512; i += 128) dst[i] = src[i];
    }
    // stage W tile: 128 (K) x 64 (N)
    for (int i = tid; i < 2048; i += 128) {
        int kk = i >> 4, j = i & 15;
        *(float4*)&wt[kk * 64 + j * 4] = *(const float4*)&W[(size_t)kk * ldW + n0 + j * 4];
    }
    __syncthreads();

    const int wid  = tid >> 5;
    const int lane = tid & 31;
    const int m    = lane & 15;                // A row (M)
    const int kb   = (lane >> 4) << 1;         // K sub-offset: lanes 0-15 -> {0,1}, 16-31 -> {2,3}
    const int ncol = (wid << 4) + (lane & 15); // column within 64-wide tile

    v8f acc = {};
    #pragma unroll
    for (int ks = 0; ks < 32; ++ks) {          // K = 128 = 32 x 4
        const int kbase = ks * 4 + kb;
        v2f a, b;
        a.x = xs[m * 128 + kbase];
        a.y = xs[m * 128 + kbase + 1];
        b.x = wt[(kbase    ) * 64 + ncol];
        b.y = wt[(kbase + 1) * 64 + ncol];
        acc = __builtin_amdgcn_wmma_f32_16x16x4_f32(
                  false, a, false, b, (short)0, acc, false, false);
    }

    // C/D layout: VGPR r -> rows {r, r+8}; lanes 0-15 rows 0..7, lanes 16-31 rows 8..15
    const int   gcol  = n0 + ncol;
    const float bb    = bias[gcol];
    const int   rbase = row0 + ((lane >> 4) << 3);
    #pragma unroll
    for (int r = 0; r < 8; ++r)
        out[(size_t)(rbase + r) * ldOut + gcol] = acc[r] + bb;
}

// ---------------------------------------------------------------------------
// K2: per-edge attention logits + segment max.
//   One wave32 per 4 edges; lane L covers channels [32L, 32L+32) (one head per
//   4-lane group). e = edge_attr @ We recomputed on the fly from TDM-staged We.
// ---------------------------------------------------------------------------
__global__ __launch_bounds__(128)
void k2_edge_alpha(const float* __restrict__ edge_attr, const int* __restrict__ ei,
                   const float* __restrict__ We,
                   const float* __restrict__ q, const float* __restrict__ k,
                   float* __restrict__ alphaBuf, unsigned* __restrict__ mEnc)
{
    __shared__ float WeS[F_EDGE * HC];      // 64 KB
    const int tid = threadIdx.x;
    stage_We_tdm(We, WeS, tid);

    const int wid  = tid >> 5;
    const int lane = tid & 31;
    const int c0   = lane << 5;
    const int ebase = blockIdx.x * 16 + wid * 4;

    for (int it = 0; it < 4; ++it) {
        const int e    = ebase + it;
        const int srcN = ei[e];
        const int dstN = ei[N_EDGES + e];

        float ea[F_EDGE];
        #pragma unroll
        for (int f = 0; f < F_EDGE; ++f) ea[f] = edge_attr[(size_t)e * F_EDGE + f];

        const float* qrow = q + (size_t)dstN * HC;
        const float* krow = k + (size_t)srcN * HC;

        float acc = 0.f;
        for (int j = 0; j < 32; ++j) {
            const int c = c0 + j;
            float ec = 0.f;
            #pragma unroll
            for (int f = 0; f < F_EDGE; ++f) ec = fmaf(ea[f], WeS[f * HC + c], ec);
            acc = fmaf(qrow[c], krow[c] + ec, acc);
        }
        // reduce the 4 lanes of each head
        acc += __shfl_xor(acc, 1);
        acc += __shfl_xor(acc, 2);

        if ((lane & 3) == 0) {
            const int h = lane >> 2;
            const float al = acc * 0.08838834764831845f;  // 1/sqrt(128)
            alphaBuf[(size_t)e * HEADS + h] = al;
            // order-preserving uint encoding for float atomicMax
            unsigned s = __float_as_uint(al);
            unsigned key = (s & 0x80000000u) ? ~s : (s | 0x80000000u);
            atomicMax(&mEnc[(size_t)dstN * HEADS + h], key);
        }
    }
}

// ---------------------------------------------------------------------------
// K3: a = exp(alpha - m); s[dst,h] += a   (alpha buffer updated in place)
// ---------------------------------------------------------------------------
__global__ void k3_softmax_norm(const int* __restrict__ ei,
                                const unsigned* __restrict__ mEnc,
                                float* __restrict__ alphaBuf,
                                float* __restrict__ sAcc)
{
    const int i = blockIdx.x * blockDim.x + threadIdx.x;
    if (i >= N_EDGES * HEADS) return;
    const int e = i >> 3, h = i & 7;
    const int d = ei[N_EDGES + e];
    const unsigned key = mEnc[d * HEADS + h];
    const unsigned s = (key & 0x80000000u) ? (key & 0x7FFFFFFFu) : ~key;
    const float m = __uint_as_float(s);
    const float a = expf(alphaBuf[i] - m);
    alphaBuf[i] = a;
    atomicAdd(&sAcc[d * HEADS + h], a);
}

// ---------------------------------------------------------------------------
// K4: unnormalized aggregate: out_un[dst,c] += a[e,h(c)] * (v[src,c] + e_c)
//   Same wave-per-4-edges structure as K2; e recomputed from TDM-staged We.
// ---------------------------------------------------------------------------
__global__ __launch_bounds__(128)
void k4_edge_aggregate(const float* __restrict__ edge_attr, const int* __restrict__ ei,
                       const float* __restrict__ We, const float* __restrict__ v,
                       const float* __restrict__ alphaBuf, float* __restrict__ outUn)
{
    __shared__ float WeS[F_EDGE * HC];      // 64 KB
    const int tid = threadIdx.x;
    stage_We_tdm(We, WeS, tid);

    const int wid  = tid >> 5;
    const int lane = tid & 31;
    const int c0   = lane << 5;
    const int h    = lane >> 2;
    const int ebase = blockIdx.x * 16 + wid * 4;

    for (int it = 0; it < 4; ++it) {
        const int e    = ebase + it;
        const int srcN = ei[e];
        const int dstN = ei[N_EDGES + e];

        float ea[F_EDGE];
        #pragma unroll
        for (int f = 0; f < F_EDGE; ++f) ea[f] = edge_attr[(size_t)e * F_EDGE + f];

        const float aV = alphaBuf[(size_t)e * HEADS + h];
        const float* vrow = v + (size_t)srcN * HC;
        float* orow = outUn + (size_t)dstN * HC;

        for (int j = 0; j < 32; ++j) {
            const int c = c0 + j;
            float ec = 0.f;
            #pragma unroll
            for (int f = 0; f < F_EDGE; ++f) ec = fmaf(ea[f], WeS[f * HC + c], ec);
            atomicAdd(&orow[c], aV * (vrow[c] + ec));
        }
    }
}

// ---------------------------------------------------------------------------
// K5: normalize, head-average, +skip, relu, node-sum -> pooled[128]
// ---------------------------------------------------------------------------
__global__ __launch_bounds__(128)
void k5_finalize(const float* __restrict__ outUn, const float* __restrict__ sAcc,
                 const float* __restrict__ skip, float* __restrict__ pooled)
{
    const int c  = threadIdx.x;
    const int n0 = blockIdx.x * 100;
    float local = 0.f;
    for (int n = n0; n < n0 + 100; ++n) {
        float acc = 0.f;
        #pragma unroll
        for (int h = 0; h < HEADS; ++h)
            acc += outUn[(size_t)n * HC + h * OUT_CH + c] / (sAcc[n * HEADS + h] + 1e-16f);
        acc = 0.125f * acc + skip[(size_t)n * OUT_CH + c];
        local += fmaxf(acc, 0.f);
    }
    atomicAdd(&pooled[c], local);
}

// ---------------------------------------------------------------------------
// K6: d_out[0] = pooled . Wd + bd
// ---------------------------------------------------------------------------
__global__ void k6_final_dot(const float* __restrict__ pooled,
                             const float* __restrict__ Wd,
                             const float* __restrict__ bd,
                             float* __restrict__ out)
{
    __shared__ float red[128];
    const int t = threadIdx.x;
    red[t] = pooled[t] * Wd[t];
    __syncthreads();
    for (int s = 64; s > 0; s >>= 1) {
        if (t < s) red[t] += red[t + s];
        __syncthreads();
    }
    if (t == 0) out[0] = red[0] + bd[0];
}

// ---------------------------------------------------------------------------
// Launch
// ---------------------------------------------------------------------------
extern "C" void kernel_launch(void* const* d_in, const int* in_sizes, int n_in,
                              void* d_out, int out_size, void* d_ws, size_t ws_size,
                              hipStream_t stream) {
    const float* x         = (const float*)d_in[0];
    const float* edge_attr = (const float*)d_in[1];
    const int*   ei        = (const int*)  d_in[2];
    const float* Wq        = (const float*)d_in[3];
    const float* bq        = (const float*)d_in[4];
    const float* Wk        = (const float*)d_in[5];
    const float* bk        = (const float*)d_in[6];
    const float* Wv        = (const float*)d_in[7];
    const float* bv        = (const float*)d_in[8];
    const float* We        = (const float*)d_in[9];
    const float* Wskip     = (const float*)d_in[10];
    const float* bskip     = (const float*)d_in[11];
    const float* Wd        = (const float*)d_in[12];
    const float* bd        = (const float*)d_in[13];

    float* w       = (float*)d_ws;
    float* qbuf    = w + Q_OFF;
    float* kbuf    = w + K_OFF;
    float* vbuf    = w + V_OFF;
    float* skipb   = w + SKIP_OFF;
    float* alphab  = w + ALPHA_OFF;
    unsigned* mEnc = (unsigned*)(w + MENC_OFF);
    float* sAcc    = w + S_OFF;
    float* outUn   = w + OUTU_OFF;
    float* pooled  = w + POOL_OFF;

    k0_zero<<<(ZERO_N4 + 255) / 256, 256, 0, stream>>>((float4*)(w + MENC_OFF), ZERO_N4);

    k1_qkvskip_gemm<<<dim3(625, 50), 128, 0, stream>>>(
        x, Wq, bq, Wk, bk, Wv, bv, Wskip, bskip, qbuf, kbuf, vbuf, skipb);

    k2_edge_alpha<<<N_EDGES / 16, 128, 0, stream>>>(
        edge_attr, ei, We, qbuf, kbuf, alphab, mEnc);

    k3_softmax_norm<<<(N_EDGES * HEADS + 255) / 256, 256, 0, stream>>>(
        ei, mEnc, alphab, sAcc);

    k4_edge_aggregate<<<N_EDGES / 16, 128, 0, stream>>>(
        edge_attr, ei, We, vbuf, alphab, outUn);

    k5_finalize<<<100, 128, 0, stream>>>(outUn, sAcc, skipb, pooled);

    k6_final_dot<<<1, 128, 0, stream>>>(pooled, Wd, bd, (float*)d_out);
}